// GNN_77386720739718
// MI455X (gfx1250) — compile-verified
//
#include <hip/hip_runtime.h>
#include <hip/hip_bf16.h>

#define EMB 128
#define NF  7
#define EF  5
#define LAYERS 5
#define BN_EPS 1e-5f

typedef __attribute__((ext_vector_type(2))) float v2f;
typedef __attribute__((ext_vector_type(8))) float v8f;

// ---------------------------------------------------------------------------
// h = x @ W_x + sum_f b_x[f]   (N x 128). One block per node, 128 threads.
// ---------------------------------------------------------------------------
__global__ __launch_bounds__(EMB)
void k_init_embed(const float* __restrict__ x, const float* __restrict__ Wx,
                  const float* __restrict__ bx, float* __restrict__ h, int N) {
    int n = blockIdx.x;
    if (n >= N) return;
    int c = threadIdx.x;
    float acc = 0.f;
#pragma unroll
    for (int f = 0; f < NF; ++f)
        acc += x[(size_t)n * NF + f] * Wx[f * EMB + c] + bx[f * EMB + c];
    h[(size_t)n * EMB + c] = acc;
}

// ---------------------------------------------------------------------------
// sl[c] = sum_f b_e[l][f][c];  also zero BN accumulators. 128 threads.
// ---------------------------------------------------------------------------
__global__ __launch_bounds__(EMB)
void k_layer_prep(const float* __restrict__ be_l, float* __restrict__ sl,
                  float* __restrict__ ssum, float* __restrict__ ssq) {
    int c = threadIdx.x;
    float acc = 0.f;
#pragma unroll
    for (int f = 0; f < EF; ++f) acc += be_l[f * EMB + c];
    sl[c]   = acc;
    ssum[c] = 0.f;
    ssq[c]  = 0.f;
}

// ---------------------------------------------------------------------------
// agg = h + sl  (vectorized float4)
// ---------------------------------------------------------------------------
__global__ __launch_bounds__(256)
void k_init_agg(const float* __restrict__ h, const float* __restrict__ sl,
                float* __restrict__ agg, int N) {
    size_t i4 = (size_t)blockIdx.x * blockDim.x + threadIdx.x;   // float4 index
    size_t total = (size_t)N * (EMB / 4);
    if (i4 >= total) return;
    int c4 = (int)(i4 & (EMB / 4 - 1));
    float4 hv = reinterpret_cast<const float4*>(h)[i4];
    float4 sv = reinterpret_cast<const float4*>(sl)[c4];
    float4 o  = make_float4(hv.x + sv.x, hv.y + sv.y, hv.z + sv.z, hv.w + sv.w);
    reinterpret_cast<float4*>(agg)[i4] = o;
}

// ---------------------------------------------------------------------------
// Edge scatter: one wave32 per edge; lane owns 4 consecutive channels.
// msg = h[src] + (edge_attr @ W_e + sl);  agg[dst] += msg  (f32 atomics -> L2)
// ---------------------------------------------------------------------------
__global__ __launch_bounds__(256)
void k_edge_scatter(const float* __restrict__ eattr, const int* __restrict__ eidx,
                    const float* __restrict__ h, const float* __restrict__ We_l,
                    const float* __restrict__ sl, float* __restrict__ agg,
                    int E, int N) {
    int tid  = threadIdx.x;
    int wave = tid >> 5, lane = tid & 31;
    long e = (long)blockIdx.x * 8 + wave;
    if (e >= E) return;
    int src = eidx[e];
    int dst = eidx[(size_t)E + e];
    int c = lane * 4;

    float4 acc = reinterpret_cast<const float4*>(sl)[lane];
#pragma unroll
    for (int f = 0; f < EF; ++f) {
        float a = eattr[(size_t)e * EF + f];            // wave-uniform
        float4 w = reinterpret_cast<const float4*>(We_l + f * EMB)[lane];
        acc.x += a * w.x; acc.y += a * w.y; acc.z += a * w.z; acc.w += a * w.w;
    }
    float4 hv = reinterpret_cast<const float4*>(h + (size_t)src * EMB)[lane];
    float* dp = agg + (size_t)dst * EMB + c;
    atomicAdd(dp + 0, hv.x + acc.x);
    atomicAdd(dp + 1, hv.y + acc.y);
    atomicAdd(dp + 2, hv.z + acc.z);
    atomicAdd(dp + 3, hv.w + acc.w);
}

// ---------------------------------------------------------------------------
// fp32 WMMA GEMM:  C[M x Nout] = act(A[M x K] @ W[K x Nout] + bias)
// Block = 256 threads (8 waves). Block tile = 32 rows x 64 cols.
// Wave w: row-tile rt = w&1 (2 x 16 rows), col-tile ct = w>>1 (4 x 16 cols).
// Both A (32 x K) and B (64 cols x K, transposed) staged in LDS with +1 row
// padding (bank = (row+k)%64, conflict-free across the 16 lanes of a half).
// K is a template parameter so the wmma loop fully unrolls and every LDS
// fragment read becomes ds_load_2addr_b32 with immediate offsets.
// Uses V_WMMA_F32_16X16X4_F32, fp32 accumulate (matches reference precision).
// ---------------------------------------------------------------------------
template <int K, bool RELU>
__global__ __launch_bounds__(256)
void k_gemm_wmma(const float* __restrict__ A, const float* __restrict__ W,
                 const float* __restrict__ bias, float* __restrict__ C,
                 int M, int Nout) {
    constexpr int KP = K + 1;                          // padded LDS row stride
    __shared__ float sA[32 * KP];
    __shared__ float sB[64 * KP];

    int tid  = threadIdx.x;
    int wave = tid >> 5, lane = tid & 31;
    int rt = wave & 1;                                 // 2 row tiles
    int ct = wave >> 1;                                // 4 col tiles
    int mBlock   = blockIdx.x * 32;
    int colBase0 = blockIdx.y * 64;

    // ---- cooperative stage of A strip (zero-padded past M), float4 reads ----
    for (int i4 = tid; i4 < 32 * K / 4; i4 += 256) {
        int idx = i4 * 4;
        int r = idx / K, kk = idx - r * K;
        int m = mBlock + r;
        float4 v = make_float4(0.f, 0.f, 0.f, 0.f);
        if (m < M) v = reinterpret_cast<const float4*>(A + (size_t)m * K + kk)[0];
        float* p = sA + r * KP + kk;
        p[0] = v.x; p[1] = v.y; p[2] = v.z; p[3] = v.w;
    }
    // ---- cooperative transposed stage of weight panel: sB[col][k] ----
    for (int i = tid; i < 64 * K; i += 256) {
        int c = i & 63, k = i >> 6;                    // coalesced global read
        sB[c * KP + k] = W[(size_t)k * Nout + colBase0 + c];
    }
    __syncthreads();

    int half = lane >> 4;                              // 0 | 1
    const float* aBase = sA + ((lane & 15) + rt * 16) * KP + half * 2;
    const float* bBase = sB + (ct * 16 + (lane & 15)) * KP + half * 2;

    v8f acc = {};
#pragma unroll
    for (int k0 = 0; k0 < K; k0 += 4) {                // imm LDS offsets k0,k0+1
        v2f a; a.x = aBase[k0]; a.y = aBase[k0 + 1];
        v2f b; b.x = bBase[k0]; b.y = bBase[k0 + 1];
        acc = __builtin_amdgcn_wmma_f32_16x16x4_f32(
            false, a, false, b, (short)0, acc, false, false);
    }

    int ncol = colBase0 + ct * 16 + (lane & 15);
    float bb = bias[ncol];
#pragma unroll
    for (int r = 0; r < 8; ++r) {
        int m = mBlock + rt * 16 + r + half * 8;       // C/D layout: M = r | r+8
        if (m < M) {
            float v = acc[r] + bb;
            if (RELU) v = v > 0.f ? v : 0.f;
            C[(size_t)m * Nout + ncol] = v;
        }
    }
}

// ---------------------------------------------------------------------------
// BN partial stats: block of 128 threads (1 per column) sweeps a row chunk,
// coalesced, then two per-column atomics.
// ---------------------------------------------------------------------------
__global__ __launch_bounds__(EMB)
void k_bn_stats(const float* __restrict__ z, float* __restrict__ ssum,
                float* __restrict__ ssq, int N) {
    int c = threadIdx.x;
    int r0 = blockIdx.x * 128;
    int r1 = min(r0 + 128, N);
    float s = 0.f, s2 = 0.f;
    for (int r = r0; r < r1; ++r) {
        float v = z[(size_t)r * EMB + c];
        s += v; s2 += v * v;
    }
    atomicAdd(&ssum[c], s);
    atomicAdd(&ssq[c], s2);
}

__global__ __launch_bounds__(EMB)
void k_bn_finalize(const float* __restrict__ ssum, const float* __restrict__ ssq,
                   float* __restrict__ mu, float* __restrict__ rinv, int N) {
    int c = threadIdx.x;
    float invN = 1.f / (float)N;
    float m = ssum[c] * invN;
    float var = ssq[c] * invN - m * m;
    mu[c]   = m;
    rinv[c] = rsqrtf(var + BN_EPS);
}

// ---------------------------------------------------------------------------
// h_out = maybe_elu((z - mu) * rinv * gamma + beta)
// ---------------------------------------------------------------------------
__global__ __launch_bounds__(256)
void k_bn_elu(const float* __restrict__ z, const float* __restrict__ mu,
              const float* __restrict__ rinv, const float* __restrict__ gamma,
              const float* __restrict__ beta, float* __restrict__ out,
              int N, int doElu) {
    size_t i = (size_t)blockIdx.x * blockDim.x + threadIdx.x;
    if (i >= (size_t)N * EMB) return;
    int c = (int)(i & (EMB - 1));
    float v = (z[i] - mu[c]) * rinv[c] * gamma[c] + beta[c];
    if (doElu) v = v > 0.f ? v : expm1f(v);
    out[i] = v;
}

// ---------------------------------------------------------------------------
extern "C" void kernel_launch(void* const* d_in, const int* in_sizes, int n_in,
                              void* d_out, int out_size, void* d_ws, size_t ws_size,
                              hipStream_t stream) {
    const float* x     = (const float*)d_in[0];   // [N,7]
    const float* eattr = (const float*)d_in[1];   // [E,5]
    const int*   eidx  = (const int*)  d_in[2];   // [2,E]
    const float* Wx    = (const float*)d_in[3];   // [7,128]
    const float* bx    = (const float*)d_in[4];   // [7,128]
    const float* We    = (const float*)d_in[5];   // [5,5,128]
    const float* be    = (const float*)d_in[6];   // [5,5,128]
    const float* W1    = (const float*)d_in[7];   // [5,128,256]
    const float* b1    = (const float*)d_in[8];   // [5,256]
    const float* W2    = (const float*)d_in[9];   // [5,256,128]
    const float* b2    = (const float*)d_in[10];  // [5,128]
    const float* gamma = (const float*)d_in[11];  // [5,128]
    const float* beta  = (const float*)d_in[12];  // [5,128]
    float* out = (float*)d_out;

    const int N = in_sizes[0] / NF;
    const int E = in_sizes[1] / EF;

    float* ws   = (float*)d_ws;
    float* h    = ws;                       // N*128
    float* agg  = h   + (size_t)N * EMB;    // N*128
    float* z    = agg + (size_t)N * EMB;    // N*128
    float* z1   = z   + (size_t)N * EMB;    // N*256
    float* sl   = z1  + (size_t)N * 2 * EMB;
    float* ssum = sl   + EMB;
    float* ssq  = ssum + EMB;
    float* mu   = ssq  + EMB;
    float* rinv = mu   + EMB;

    k_init_embed<<<N, EMB, 0, stream>>>(x, Wx, bx, h, N);

    const int elemBlocks = (int)(((size_t)N * EMB + 255) / 256);
    const int aggBlocks  = (int)(((size_t)N * (EMB / 4) + 255) / 256);
    const int edgeBlocks = (E + 7) / 8;
    const int gemmRows   = (N + 31) / 32;
    const int bnBlocks   = (N + 127) / 128;

    for (int l = 0; l < LAYERS; ++l) {
        k_layer_prep<<<1, EMB, 0, stream>>>(be + (size_t)l * EF * EMB, sl, ssum, ssq);
        k_init_agg<<<aggBlocks, 256, 0, stream>>>(h, sl, agg, N);
        k_edge_scatter<<<edgeBlocks, 256, 0, stream>>>(
            eattr, eidx, h, We + (size_t)l * EF * EMB, sl, agg, E, N);

        // z1 = relu(agg @ W1[l] + b1[l])   [N x 256]
        k_gemm_wmma<EMB, true><<<dim3(gemmRows, 256 / 64), 256, 0, stream>>>(
            agg, W1 + (size_t)l * EMB * 2 * EMB, b1 + (size_t)l * 2 * EMB,
            z1, N, 2 * EMB);
        // z = z1 @ W2[l] + b2[l]           [N x 128]
        k_gemm_wmma<2 * EMB, false><<<dim3(gemmRows, EMB / 64), 256, 0, stream>>>(
            z1, W2 + (size_t)l * 2 * EMB * EMB, b2 + (size_t)l * EMB,
            z, N, EMB);

        k_bn_stats<<<bnBlocks, EMB, 0, stream>>>(z, ssum, ssq, N);
        k_bn_finalize<<<1, EMB, 0, stream>>>(ssum, ssq, mu, rinv, N);

        int last = (l == LAYERS - 1);
        k_bn_elu<<<elemBlocks, 256, 0, stream>>>(
            z, mu, rinv, gamma + (size_t)l * EMB, beta + (size_t)l * EMB,
            last ? out : h, N, !last);
    }
}